// Coord_fine_68968584839907
// MI455X (gfx1250) — compile-verified
//
#include <hip/hip_runtime.h>
#include <hip/hip_bf16.h>
#include <stdint.h>

typedef __attribute__((ext_vector_type(2))) float v2f;
typedef __attribute__((ext_vector_type(8))) float v8f;

namespace {
constexpr int NBATCH = 32;
constexpr int NPT    = 4;
constexpr int NCH    = 33;
constexpr int IMHW   = 256;
constexpr int NPATCH = 36;
constexpr int KTOT   = NCH * 64;   // 2112 = 33*8*8
constexpr int KCHUNK = 528;        // KTOT / 4, multiple of 4
constexpr int MROWS  = NBATCH * NPATCH; // 1152
constexpr int FEAT   = 107;
}

// ---------------------------------------------------------------------------
// threefry2x32 (JAX default PRNG), 20 rounds, exact key-injection schedule
// ---------------------------------------------------------------------------
__device__ __forceinline__ void threefry2x32(uint32_t k0, uint32_t k1,
                                             uint32_t x0, uint32_t x1,
                                             uint32_t& o0, uint32_t& o1) {
  const uint32_t k2 = k0 ^ k1 ^ 0x1BD11BDAu;
#define TF_ROT(v, r) (((v) << (r)) | ((v) >> (32 - (r))))
#define TF_R4(a, b, c, dd)                                                    \
  x0 += x1; x1 = TF_ROT(x1, a); x1 ^= x0;                                     \
  x0 += x1; x1 = TF_ROT(x1, b); x1 ^= x0;                                     \
  x0 += x1; x1 = TF_ROT(x1, c); x1 ^= x0;                                     \
  x0 += x1; x1 = TF_ROT(x1, dd); x1 ^= x0;
  x0 += k0; x1 += k1;
  TF_R4(13, 15, 26, 6)  x0 += k1; x1 += k2 + 1u;
  TF_R4(17, 29, 16, 24) x0 += k2; x1 += k0 + 2u;
  TF_R4(13, 15, 26, 6)  x0 += k0; x1 += k1 + 3u;
  TF_R4(17, 29, 16, 24) x0 += k1; x1 += k2 + 4u;
  TF_R4(13, 15, 26, 6)  x0 += k2; x1 += k0 + 5u;
  o0 = x0; o1 = x1;
#undef TF_R4
#undef TF_ROT
}

// bernoulli(key, 0.9, (32,4,107)) mask value (already divided by keep-prob)
__device__ __forceinline__ float drop_mask(uint32_t k0, uint32_t k1, uint32_t e) {
  const uint32_t HALFN = 6848u;  // (32*4*107)/2
  uint32_t i = (e < HALFN) ? e : (e - HALFN);
  uint32_t h0, h1;
  threefry2x32(k0, k1, i, i + HALFN, h0, h1);
  uint32_t bits = (e < HALFN) ? h0 : h1;
  float u = __uint_as_float((bits >> 9) | 0x3f800000u) - 1.0f;
  return (u < 0.9f) ? (1.0f / 0.9f) : 0.0f;
}

// ---------------------------------------------------------------------------
// Kernel A: replay the i*j indexed in-place shift accumulation; emit the
// 36 crop centers (in step order) and the final lh table.
// ---------------------------------------------------------------------------
__global__ void k_centers(const float* __restrict__ x,
                          int* __restrict__ centers,
                          float* __restrict__ lh_out) {
  int b = threadIdx.x;
  if (b >= NBATCH) return;
  const float S0[9] = {-8.f/255.f, -8.f/255.f, 0.f, 8.f/255.f, 8.f/255.f,
                        8.f/255.f, 0.f, -8.f/255.f, 0.f};
  const float S1[9] = {0.f, 8.f/255.f, 8.f/255.f, 8.f/255.f, 0.f,
                       -8.f/255.f, 8.f/255.f, -8.f/255.f, 0.f};
  float lh0[NPATCH], lh1[NPATCH];
  for (int n = 0; n < NPATCH; ++n) {
    lh0[n] = x[(b * NPT + n / 9) * 2 + 0];
    lh1[n] = x[(b * NPT + n / 9) * 2 + 1];
  }
  int k = 0;
  for (int i = 0; i < NPT; ++i) {
    for (int j = 0; j < 9; ++j, ++k) {
      int idx = i * j;  // reproduces the reference's indexing bug
      lh0[idx] += S0[j];
      lh1[idx] += S1[j];
      float fx = rintf(lh0[idx] * 255.f);              // round-half-even
      float fy = rintf((lh1[idx] + 1.f) * 255.f);
      fx = fminf(fmaxf(fx, 0.f), 255.f);
      fy = fminf(fmaxf(fy, 0.f), 255.f);
      centers[(b * NPATCH + k) * 2 + 0] = (int)fx;
      centers[(b * NPATCH + k) * 2 + 1] = (int)fy;
    }
  }
  for (int n = 0; n < NPATCH; ++n) {
    lh_out[(b * NPATCH + n) * 2 + 0] = lh0[n];
    lh_out[(b * NPATCH + n) * 2 + 1] = lh1[n];
  }
}

// ---------------------------------------------------------------------------
// Kernel B: patch_feat GEMM [1152 x 2112] x [2112 x 33] via f32 WMMA 16x16x4.
// One block = 16 M-rows. LDS holds a 16 x 528 K-chunk of gathered crop data.
// Waves 0..2 each own a 16-wide N tile (33 channels padded to 48).
// ---------------------------------------------------------------------------
__global__ __launch_bounds__(128) void k_patch_wmma(
    const float* __restrict__ dimg, const float* __restrict__ conv_w,
    const float* __restrict__ conv_b, const int* __restrict__ centers,
    float* __restrict__ pf) {
  __shared__ float lds_a[16 * KCHUNK];   // 33 KB
  __shared__ int s_ix[16], s_iy[16], s_b[16];
  const int tid = threadIdx.x;
  const int m_base = blockIdx.x * 16;
  if (tid < 16) {
    int m = m_base + tid;
    int bb = m / NPATCH;
    int n = m % NPATCH;
    s_b[tid]  = bb;
    s_ix[tid] = centers[(bb * NPATCH + n) * 2 + 0];
    s_iy[tid] = centers[(bb * NPATCH + n) * 2 + 1];
  }
  const int wave = tid >> 5;
  const int lane = tid & 31;
  const int ncol = wave * 16 + (lane & 15);       // output channel (0..47)
  const int ncc  = (ncol > 32) ? 32 : ncol;        // clamp for OOB-safe B loads
  const int koff = (lane < 16) ? 0 : 2;            // A/B frag K sub-offset
  const int rloc = lane & 15;                      // A-matrix row (M)
  v8f acc = {};
  for (int kc = 0; kc < KTOT; kc += KCHUNK) {
    __syncthreads();
    // cooperatively gather crop chunk: k = c*64 + h*8 + w
    for (int e = tid; e < 16 * KCHUNK; e += 128) {
      int row = e / KCHUNK;
      int kk  = e - row * KCHUNK;
      int k   = kc + kk;
      int c = k >> 6, h = (k >> 3) & 7, w = k & 7;
      int r  = s_ix[row] + h - 4;
      int cc = s_iy[row] + w - 4;
      float v = 0.0f;
      if ((unsigned)r < (unsigned)IMHW && (unsigned)cc < (unsigned)IMHW)
        v = dimg[(((size_t)s_b[row] * NCH + c) * IMHW + r) * IMHW + cc];
      lds_a[row * KCHUNK + kk] = v;
    }
    __syncthreads();
    if (wave < 3) {  // wave-uniform branch: EXEC all-ones inside
      const float* bp = conv_w + (size_t)ncc * KTOT + kc + koff;
#pragma unroll 4
      for (int kk = 0; kk < KCHUNK; kk += 4) {
        // A frag: lanes 0-15 -> (M=lane, K=kk,kk+1); lanes 16-31 -> K=kk+2,kk+3
        v2f a = *reinterpret_cast<const v2f*>(&lds_a[rloc * KCHUNK + kk + koff]);
        // B frag: B[k][n] = conv_w[n][k] -> contiguous pair per lane
        v2f bm = *reinterpret_cast<const v2f*>(bp + kk);
        acc = __builtin_amdgcn_wmma_f32_16x16x4_f32(
            false, a, false, bm, (short)0, acc, false, false);
      }
    }
  }
  if (wave < 3 && ncol < NCH) {
    const float bias = conv_b[ncol];
    // D layout: VGPR r = row (M=r | M=r+8), lanes 0-15 | 16-31, N = lane%16
    const int mrow0 = m_base + ((lane < 16) ? 0 : 8);
#pragma unroll
    for (int r = 0; r < 8; ++r)
      pf[(size_t)(mrow0 + r) * NCH + ncol] = acc[r] + bias;
  }
}

// ---------------------------------------------------------------------------
// Kernel C: hypergraph message passing + dropout + heads. One thread per
// (batch, point). All sizes are tiny ([32,4,107]).
// ---------------------------------------------------------------------------
__global__ __launch_bounds__(128) void k_head(
    const float* __restrict__ x, const float* __restrict__ H,
    const float* __restrict__ T, const float* __restrict__ Wt,
    const float* __restrict__ l1w, const float* __restrict__ l1b,
    const float* __restrict__ l2w, const float* __restrict__ l2b,
    const float* __restrict__ pf, const float* __restrict__ lh,
    float* __restrict__ out) {
  __shared__ float smsg[128][FEAT];   // ~54.8 KB, each thread uses its own row
  const int t = threadIdx.x;
  const int b = t >> 2;
  const int p = t & 3;
  // M1[p,n] = sum_h (W[h] * sum_m T[p,m]*H[m,h]) * H[n,h]
  float A1[16];
#pragma unroll
  for (int h = 0; h < 16; ++h) {
    float s = 0.f;
    for (int m = 0; m < NPATCH; ++m) s += T[p * NPATCH + m] * H[m * 16 + h];
    A1[h] = s * Wt[h];
  }
  float M1[NPATCH];
  for (int n = 0; n < NPATCH; ++n) {
    float s = 0.f;
#pragma unroll
    for (int h = 0; h < 16; ++h) s += A1[h] * H[n * 16 + h];
    M1[n] = s;
  }
  const float* lhb = lh + b * NPATCH * 2;
  const float* pfb = pf + (size_t)b * NPATCH * NCH;
  // message[f] = sum_n M1[n] * gin[b,n,f]
  for (int f = 0; f < FEAT; ++f) {
    float s = 0.f;
    if (f < NCH) {
      for (int n = 0; n < NPATCH; ++n) s += M1[n] * pfb[n * NCH + f];
    } else if (f < NCH + 72) {
      int j = (f - NCH) >> 1, c = (f - NCH) & 1;
      float lj = lhb[j * 2 + c];
      for (int n = 0; n < NPATCH; ++n) s += M1[n] * (lj - lhb[n * 2 + c]);
    } else {
      int c = f - (NCH + 72);
      for (int n = 0; n < NPATCH; ++n) s += M1[n] * lhb[n * 2 + c];
    }
    smsg[t][f] = s;
  }
  // k1, k2 = split(key(42)); key data = (0, 42)
  uint32_t y00, y10, y01, y11;
  threefry2x32(0u, 42u, 0u, 2u, y00, y10);
  threefry2x32(0u, 42u, 1u, 3u, y01, y11);
  // k1 = (y00, y01), k2 = (y10, y11)
  float z1[2] = {l1b[0], l1b[1]};
  float z2[9];
#pragma unroll
  for (int o = 0; o < 9; ++o) z2[o] = l2b[o];
  const uint32_t ebase = (uint32_t)t * FEAT;
  for (int f = 0; f < FEAT; ++f) {
    float msg = smsg[t][f];
    float x1 = msg * drop_mask(y00, y01, ebase + (uint32_t)f);
    float x2 = msg * drop_mask(y10, y11, ebase + (uint32_t)f);
    z1[0] += x1 * l1w[f];
    z1[1] += x1 * l1w[FEAT + f];
#pragma unroll
    for (int o = 0; o < 9; ++o) z2[o] += x2 * l2w[o * FEAT + f];
  }
  const float off0 = 1.f / (1.f + __expf(-z1[0]));
  const float off1 = 1.f / (1.f + __expf(-z1[1]));
  float mx = z2[0];
#pragma unroll
  for (int o = 1; o < 9; ++o) mx = fmaxf(mx, z2[o]);
  float den = 0.f, e9[9];
#pragma unroll
  for (int o = 0; o < 9; ++o) { e9[o] = __expf(z2[o] - mx); den += e9[o]; }
  const float S0[9] = {-8.f/255.f, -8.f/255.f, 0.f, 8.f/255.f, 8.f/255.f,
                        8.f/255.f, 0.f, -8.f/255.f, 0.f};
  const float S1[9] = {0.f, 8.f/255.f, 8.f/255.f, 8.f/255.f, 0.f,
                       -8.f/255.f, 8.f/255.f, -8.f/255.f, 0.f};
  float sv0 = 0.f, sv1 = 0.f;
#pragma unroll
  for (int o = 0; o < 9; ++o) {
    float dv = e9[o] / den;
    sv0 += dv * S0[o];
    sv1 += dv * S1[o];
  }
  const int xi = (b * NPT + p) * 2;
  out[xi + 0] = (x[xi + 0] + off0 * sv0) * 255.f;
  out[xi + 1] = (x[xi + 1] + off1 * sv1) * 255.f;
}

// ---------------------------------------------------------------------------
extern "C" void kernel_launch(void* const* d_in, const int* in_sizes, int n_in,
                              void* d_out, int out_size, void* d_ws, size_t ws_size,
                              hipStream_t stream) {
  (void)in_sizes; (void)n_in; (void)out_size; (void)ws_size;
  const float* dimg   = (const float*)d_in[0];
  const float* x      = (const float*)d_in[1];
  const float* conv_w = (const float*)d_in[2];
  const float* conv_b = (const float*)d_in[3];
  const float* H      = (const float*)d_in[4];
  const float* T      = (const float*)d_in[5];
  const float* W      = (const float*)d_in[6];
  const float* l1w    = (const float*)d_in[7];
  const float* l1b    = (const float*)d_in[8];
  const float* l2w    = (const float*)d_in[9];
  const float* l2b    = (const float*)d_in[10];
  float* out = (float*)d_out;
  // workspace layout (256B-aligned sections)
  int*   centers = (int*)d_ws;                        // 32*36*2 ints  = 9216 B
  float* lh      = (float*)((char*)d_ws + 9216);      // 32*36*2 f32   = 9216 B
  float* pf      = (float*)((char*)d_ws + 18432);     // 1152*33 f32   = 152064 B

  k_centers<<<1, 32, 0, stream>>>(x, centers, lh);
  k_patch_wmma<<<MROWS / 16, 128, 0, stream>>>(dimg, conv_w, conv_b, centers, pf);
  k_head<<<1, 128, 0, stream>>>(x, H, T, W, l1w, l1b, l2w, l2b, pf, lh, out);
}